// PageTable_23811298689264
// MI455X (gfx1250) — compile-verified
//
#include <hip/hip_runtime.h>
#include <hip/hip_bf16.h>

// Page-table allocation pipeline for MI455X (gfx1250, wave32).
// Integer scatter/scan workload: no WMMA applies. ~3.5 MB total traffic ->
// latency/launch-bound; fused into 6 stream-ordered kernels; bulk copies use
// int4 (global_{load,store}_b128) to minimize VMEM instruction count.

#define CHUNK 256           // tokens / pages per block (8 wave32 waves)
#define MAX_BINS 1040       // supports S up to 1039 (reference: S=512 -> 513 bins)

__device__ __forceinline__ int fdiv_floor(int a, int b) {
    int q = a / b;
    if ((a % b != 0) && ((a ^ b) < 0)) q--;
    return q;
}
__device__ __forceinline__ int ceil_pages(int x, int page) {
    return fdiv_floor(x + page - 1, page);   // Python // semantics
}

// ---------------------------------------------------------------- k0: zero bins
__global__ void k_init(int* counts, int bins) {
    for (int i = threadIdx.x; i < bins; i += blockDim.x) counts[i] = 0;
}

// ------------------------------------- k1: token chunk histogram + free counts
__global__ void k_hist(const int* __restrict__ tok, const int* __restrict__ refc,
                       int* counts_all, int* chist, int* bfc,
                       int P, int NP, int S, int NCt, int NCp) {
    const int BINS = S + 1;
    __shared__ int sh[MAX_BINS];
    __shared__ int freeCnt;
    const int c = blockIdx.x, tid = threadIdx.x;

    if (c < NCt) {
        for (int s = tid; s < BINS; s += blockDim.x) sh[s] = 0;
        __syncthreads();
        int t = c * CHUNK + tid;
        if (t < P) {
            int v = tok[t];
            int id = (v < 0) ? S : (v > S ? S : v);
            atomicAdd(&sh[id], 1);                       // LDS atomic (ds_add)
        }
        __syncthreads();
        for (int s = tid; s < BINS; s += blockDim.x) {
            int h = sh[s];
            chist[(size_t)c * BINS + s] = h;
            if (h) atomicAdd(&counts_all[s], h);         // global atomic
        }
    }
    if (c < NCp) {
        if (tid == 0) freeCnt = 0;
        __syncthreads();
        int p = c * CHUNK + tid;
        if (p < NP && refc[p] == 0) atomicAdd(&freeCnt, 1);
        __syncthreads();
        if (tid == 0) bfc[c] = freeCnt;
    }
}

// ------------- k2: sorted-unique, new_lens, offsets, cu_q_lens, chunk prefixes
__global__ void k_scan(const int* __restrict__ counts, const int* __restrict__ seq_lens,
                       int* upd, int* offs, int* bfc, int* chist,
                       int* out_newlens, int* out_cu, int* out_numseqs,
                       int S, int NCt, int NCp, int PAGE) {
    const int BINS = S + 1;
    const int tid = threadIdx.x;

    if (tid == 0) {
        // sorted unique seq ids (ascending), padded with -1
        int k = 0;
        for (int s = 0; s <= S; s++)
            if (counts[s] > 0 && k < S) upd[k++] = s;
        out_numseqs[0] = k;                  // mask = updated_seqs >= 0 (incl. overflow id)
        for (int i = k; i < S; i++) upd[i] = -1;

        // cu_q_lens + per-slot allocation offsets (exclusive cumsum of need)
        int cu = 0, cum = 0;
        out_cu[0] = 0;
        for (int i = 0; i < S; i++) {
            int id = upd[i];
            bool inr = (id >= 0 && id < S);
            int nc = inr ? counts[id] : 0;   // overflow bucket contributes 0
            cu += nc;
            out_cu[i + 1] = cu;
            offs[i] = cum;
            if (inr) {
                int sl = seq_lens[id];
                int nl = (sl >= 0) ? (sl + counts[id]) : -1;
                int need = ceil_pages(nl, PAGE) - ceil_pages(sl, PAGE);
                if (need > 0) cum += need;
            }
        }
        // exclusive prefix over per-chunk free-page counts
        int run = 0;
        for (int c = 0; c < NCp; c++) { int t = bfc[c]; bfc[c] = run; run += t; }
    }

    // new sequence lengths (parallel)
    for (int s = tid; s < S; s += blockDim.x) {
        int sl = seq_lens[s];
        out_newlens[s] = (sl >= 0) ? (sl + counts[s]) : -1;
    }

    // per-id exclusive prefix across token chunks (parallel over bins)
    for (int id = tid; id < BINS; id += blockDim.x) {
        int run = 0;
        for (int c = 0; c < NCt; c++) {
            size_t idx = (size_t)c * BINS + id;
            int t = chist[idx];
            chist[idx] = run;
            run += t;
        }
    }
}

// ----------------- k3: compact free list + copy page_indices / ref_counts out
__global__ void k_free_copy(const int* __restrict__ refc, const int* __restrict__ pidx,
                            const int* __restrict__ bfc_prefix, int* flist,
                            int* out_pidx, int* out_refc,
                            int NP, int NCp, int PI_N) {
    const int b = blockIdx.x, tid = threadIdx.x;
    const int gsz = gridDim.x * blockDim.x;
    const int g = b * blockDim.x + tid;

    // vectorized b128 copies (reference sizes are multiples of 4)
    const int n4p = PI_N >> 2;
    const int4* src4p = (const int4*)pidx;
    int4*       dst4p = (int4*)out_pidx;
    for (int i = g; i < n4p; i += gsz) dst4p[i] = src4p[i];
    for (int i = (n4p << 2) + g; i < PI_N; i += gsz) out_pidx[i] = pidx[i];

    const int n4r = NP >> 2;
    const int4* src4r = (const int4*)refc;
    int4*       dst4r = (int4*)out_refc;
    for (int i = g; i < n4r; i += gsz) dst4r[i] = src4r[i];
    for (int i = (n4r << 2) + g; i < NP; i += gsz) out_refc[i] = refc[i];

    if (b < NCp && tid == 0) {
        int base = bfc_prefix[b];
        int start = b * CHUNK;
        int end = start + CHUNK; if (end > NP) end = NP;
        for (int p = start; p < end; p++)
            if (refc[p] == 0) flist[base++] = p;   // free pages, ascending
    }
}

// ------------------------------------------- k4: bulk page allocation scatter
__global__ void k_alloc(const int* __restrict__ upd, const int* __restrict__ offs,
                        const int* __restrict__ seq_lens, const int* __restrict__ new_lens,
                        const int* __restrict__ flist,
                        int* out_pidx, int* out_refc,
                        int S, int PPS, int PAGE) {
    const int i = blockIdx.x;
    if (i >= S) return;
    int id = upd[i];
    if (id < 0 || id >= S) return;               // overflow / padding slots: need==0
    int old_s = ceil_pages(seq_lens[id], PAGE);
    int new_s = ceil_pages(new_lens[id], PAGE);
    int off = offs[i];
    if (threadIdx.x == 0 && new_s > old_s)
        __builtin_prefetch(&flist[off], 0, 0);   // global_prefetch_b8 (gfx1250)
    for (int j = threadIdx.x; j < PPS; j += blockDim.x) {
        if (j >= old_s && j < new_s) {
            int page = flist[off + (j - old_s)];
            out_pidx[(size_t)id * PPS + j] = page;
            atomicAdd(&out_refc[page], 1);       // distinct pages; atomic for safety
        }
    }
}

// -------------------- k5: batch slices + per-token destinations (after alloc)
__global__ void k_batch_dests(const int* __restrict__ tok, const int* __restrict__ seq_lens,
                              const int* __restrict__ upd, const int* __restrict__ chist,
                              const int* __restrict__ out_pidx, const int* __restrict__ out_newlens,
                              int* out_bpidx, int* out_bslens, int* out_dests,
                              int P, int S, int PPS, int NCt, int PAGE) {
    const int BINS = S + 1;
    const int b = blockIdx.x, tid = threadIdx.x;

    if (b < S) {
        // _slice_batch_info: overflow id (==S) keeps mask true, aliases row 0 (safe_upd)
        int id = upd[b];
        bool maskv = (id >= 0);
        int safe = (id >= 0 && id < S) ? id : 0;
        if ((PPS & 3) == 0) {
            const int n4 = PPS >> 2;
            const int4* src4 = (const int4*)(out_pidx + (size_t)safe * PPS);
            int4*       dst4 = (int4*)(out_bpidx + (size_t)b * PPS);
            const int4 inv4 = make_int4(-1, -1, -1, -1);
            for (int j = tid; j < n4; j += blockDim.x)
                dst4[j] = maskv ? src4[j] : inv4;
        } else {
            for (int j = tid; j < PPS; j += blockDim.x)
                out_bpidx[(size_t)b * PPS + j] = maskv ? out_pidx[(size_t)safe * PPS + j] : -1;
        }
        if (tid == 0) out_bslens[b] = maskv ? out_newlens[safe] : -1;
        return;
    }

    const int c = b - S;
    if (c >= NCt) return;                        // uniform per block
    __shared__ int ids[CHUNK];
    int t = c * CHUNK + tid;
    int id = S;
    if (t < P) {
        int v = tok[t];
        id = (v < 0) ? S : (v > S ? S : v);
    }
    ids[tid] = id;
    __syncthreads();
    if (t < P) {
        // stable rank of token within its sequence: cross-chunk prefix + in-chunk scan
        int rank = chist[(size_t)c * BINS + id];
        for (int j = 0; j < tid; j++) rank += (ids[j] == id);
        bool tv = (id < S);
        int sa = tv ? id : 0;
        int sl = seq_lens[sa];
        int cursor = ((sl < 0) ? 0 : sl) + rank;
        int pi = cursor / PAGE;
        if (pi > PPS - 1) pi = PPS - 1;
        if (pi < 0) pi = 0;
        int po = cursor % PAGE;
        int page = out_pidx[(size_t)sa * PPS + pi];
        out_dests[t] = (page < 0 || !tv) ? -1 : page * PAGE + po;
    }
}

static inline int imax(int a, int b) { return a > b ? a : b; }

extern "C" void kernel_launch(void* const* d_in, const int* in_sizes, int n_in,
                              void* d_out, int out_size, void* d_ws, size_t ws_size,
                              hipStream_t stream) {
    const int* tok      = (const int*)d_in[0];   // token_seq_ids [P]
    const int* seq_lens = (const int*)d_in[1];   // [S]
    const int* pidx     = (const int*)d_in[2];   // [S, PPS]
    const int* refc     = (const int*)d_in[3];   // [NP]

    const int P    = in_sizes[0];
    const int S    = in_sizes[1];
    const int PI_N = in_sizes[2];
    const int PPS  = PI_N / S;
    const int NP   = in_sizes[3];
    const int BINS = S + 1;
    const int PAGE = 128;
    const int NCt  = (P + CHUNK - 1) / CHUNK;
    const int NCp  = (NP + CHUNK - 1) / CHUNK;

    // d_out layout (int32 views), in reference return order
    int* out      = (int*)d_out;
    int* o_npidx  = out;                                  // [S*PPS]
    int* o_nrefc  = o_npidx + (size_t)S * PPS;            // [NP]
    int* o_nlens  = o_nrefc + NP;                         // [S]
    int* o_bpidx  = o_nlens + S;                          // [S*PPS]
    int* o_bslens = o_bpidx + (size_t)S * PPS;            // [S]
    int* o_cu     = o_bslens + S;                         // [S+1]
    int* o_nseqs  = o_cu + (S + 1);                       // [1]
    int* o_dests  = o_nseqs + 1;                          // [P]

    // workspace layout
    int* ws       = (int*)d_ws;
    int* w_counts = ws;                                   // [BINS]
    int* w_upd    = w_counts + BINS;                      // [S]
    int* w_off    = w_upd + S;                            // [S]
    int* w_chist  = w_off + S;                            // [NCt*BINS]
    int* w_bfc    = w_chist + (size_t)NCt * BINS;         // [NCp]
    int* w_flist  = w_bfc + NCp;                          // [NP]

    k_init<<<1, 256, 0, stream>>>(w_counts, BINS);

    k_hist<<<imax(NCt, NCp), CHUNK, 0, stream>>>(tok, refc, w_counts, w_chist, w_bfc,
                                                 P, NP, S, NCt, NCp);

    k_scan<<<1, 512, 0, stream>>>(w_counts, seq_lens, w_upd, w_off, w_bfc, w_chist,
                                  o_nlens, o_cu, o_nseqs, S, NCt, NCp, PAGE);

    int nb_fc = imax(NCp, ((PI_N >> 2) + CHUNK - 1) / CHUNK);
    k_free_copy<<<nb_fc, CHUNK, 0, stream>>>(refc, pidx, w_bfc, w_flist,
                                             o_npidx, o_nrefc, NP, NCp, PI_N);

    k_alloc<<<S, CHUNK, 0, stream>>>(w_upd, w_off, seq_lens, o_nlens, w_flist,
                                     o_npidx, o_nrefc, S, PPS, PAGE);

    k_batch_dests<<<S + NCt, CHUNK, 0, stream>>>(tok, seq_lens, w_upd, w_chist,
                                                 o_npidx, o_nlens,
                                                 o_bpidx, o_bslens, o_dests,
                                                 P, S, PPS, NCt, PAGE);
}